// Model_23940147707905
// MI455X (gfx1250) — compile-verified
//
#include <hip/hip_runtime.h>
#include <hip/hip_bf16.h>

#ifndef __has_builtin
#define __has_builtin(x) 0
#endif

// CDNA5 async global->LDS copy path (ASYNCcnt-tracked). Guarded so the file
// always compiles; histogram will show whether it lowered.
#if __has_builtin(__builtin_amdgcn_global_load_async_to_lds_b32)
#define USE_ASYNC_LDS 1
#else
#define USE_ASYNC_LDS 0
#endif

#define NTYPE 9

typedef __attribute__((address_space(1))) int g1_int;   // global
typedef __attribute__((address_space(3))) int l3_int;   // LDS

struct F3 { float x, y, z; };

__device__ __forceinline__ float fast_rcp(float x)  { return __builtin_amdgcn_rcpf(x); }
__device__ __forceinline__ float fast_sqrt(float x) { return __builtin_amdgcn_sqrtf(x); }

// ---------------------------------------------------------------------------
// Node features: [N,12] = concat(velocity(3), one_hot(9)), normalized.
// One thread per node; B96 loads, 3x B128 stores (row stride 48B -> 16B aligned).
// ---------------------------------------------------------------------------
__global__ __launch_bounds__(256) void node_feat_kernel(
    const float* __restrict__ wp, const float* __restrict__ pwp,
    const int* __restrict__ ntype,
    const float* __restrict__ nmean, const float* __restrict__ nstd,
    float* __restrict__ out, int N)
{
  int n = blockIdx.x * 256 + threadIdx.x;
  if (n >= N) return;

  F3 w = *(const F3*)(wp  + 3 * (size_t)n);
  F3 p = *(const F3*)(pwp + 3 * (size_t)n);
  int t = ntype[n];

  float r[12];
  r[0] = ((w.x - p.x) - nmean[0]) * fast_rcp(nstd[0]);
  r[1] = ((w.y - p.y) - nmean[1]) * fast_rcp(nstd[1]);
  r[2] = ((w.z - p.z) - nmean[2]) * fast_rcp(nstd[2]);
#pragma unroll
  for (int j = 0; j < NTYPE; ++j) {
    float oh = (j == t) ? 1.0f : 0.0f;          // v_cmp + v_cndmask, no scratch
    r[3 + j] = (oh - nmean[3 + j]) * fast_rcp(nstd[3 + j]);
  }

  float4* o = (float4*)(out + 12 * (size_t)n);
  o[0] = make_float4(r[0], r[1], r[2],  r[3]);
  o[1] = make_float4(r[4], r[5], r[6],  r[7]);
  o[2] = make_float4(r[8], r[9], r[10], r[11]);
}

// ---------------------------------------------------------------------------
// Edge features: cell-centric. Thread = one cell; gathers its 3 vertices once
// and emits all 6 directed edges (reverse edge = sign flip, same norms).
// Edge index layout (from _two_way_edges): for pair-group g in {0,1,2}:
//   forward  edge:   e = g*C + cid        (sender = v[g],     receiver = v[g+1 mod 3])
//   backward edge:   e = 3C + g*C + cid   (roles swapped -> negated rel vectors)
// ---------------------------------------------------------------------------
__device__ __forceinline__ void write_edge(float* __restrict__ o,
    float dx, float dy, float dz, float nw, float mx, float my, float nm,
    const float* em, const float* ei)
{
  o[0] = (dx - em[0]) * ei[0];
  o[1] = (dy - em[1]) * ei[1];
  o[2] = (dz - em[2]) * ei[2];
  o[3] = (nw - em[3]) * ei[3];
  o[4] = (mx - em[4]) * ei[4];
  o[5] = (my - em[5]) * ei[5];
  o[6] = (nm - em[6]) * ei[6];
}

__device__ __forceinline__ void do_pair(float* __restrict__ eo, size_t C, size_t cid, int g,
    const F3& wa, const F3& wb, const float2& ma, const float2& mb,
    const float* em, const float* ei)
{
  float dx = wa.x - wb.x, dy = wa.y - wb.y, dz = wa.z - wb.z;
  float mx = ma.x - mb.x, my = ma.y - mb.y;
  float nw = fast_sqrt(dx * dx + dy * dy + dz * dz);
  float nm = fast_sqrt(mx * mx + my * my);
  write_edge(eo + 7 * ((size_t)g * C + cid),        dx,  dy,  dz, nw,  mx,  my, nm, em, ei);
  write_edge(eo + 7 * ((size_t)(3 + g) * C + cid), -dx, -dy, -dz, nw, -mx, -my, nm, em, ei);
}

__global__ __launch_bounds__(256) void edge_feat_kernel(
    const float* __restrict__ wp, const float* __restrict__ mp,
    const int* __restrict__ cells,
    const float* __restrict__ emean, const float* __restrict__ estd,
    float* __restrict__ eout, int C, int N)
{
  __shared__ int sc[3 * 256];                    // this block's 256 cells (3 KB)
  const int t      = threadIdx.x;
  const int cell0  = blockIdx.x * 256;
  const int base3  = cell0 * 3;
  const int total3 = C * 3;

  // Warm L2 with the gather-target arrays (per-lane cacheline prefetch,
  // lowers to global_prefetch_b8). Positions fit easily in the 192MB L2.
  {
    int pid = blockIdx.x * 256 + t;
    if (pid * 32 < 3 * N) __builtin_prefetch(wp + (size_t)pid * 32, 0, 1);
    if (pid * 32 < 2 * N) __builtin_prefetch(mp + (size_t)pid * 32, 0, 1);
  }

  // Stage the block's cell indices (768 dwords) into LDS, fully coalesced.
#if USE_ASYNC_LDS
#pragma unroll
  for (int r = 0; r < 3; ++r) {
    int dw = base3 + r * 256 + t;
    if (dw < total3) {
      __builtin_amdgcn_global_load_async_to_lds_b32(
          (g1_int*)(cells + dw),
          (l3_int*)(&sc[r * 256 + t]),
          /*offset=*/0, /*cpol=*/0);
    }
  }
#if __has_builtin(__builtin_amdgcn_s_wait_asynccnt)
  __builtin_amdgcn_s_wait_asynccnt(0);
#else
  asm volatile("s_wait_asynccnt 0" ::: "memory");
#endif
#else
#pragma unroll
  for (int r = 0; r < 3; ++r) {
    int dw = base3 + r * 256 + t;
    if (dw < total3) sc[r * 256 + t] = cells[dw];
  }
#endif
  __syncthreads();   // ASYNCcnt is per-wave; barrier publishes LDS across waves

  int cid = cell0 + t;
  if (cid >= C) return;

  int v0 = sc[3 * t + 0];
  int v1 = sc[3 * t + 1];
  int v2 = sc[3 * t + 2];

  F3 w0 = *(const F3*)(wp + 3 * (size_t)v0);     // B96 gathers (L2-resident)
  F3 w1 = *(const F3*)(wp + 3 * (size_t)v1);
  F3 w2 = *(const F3*)(wp + 3 * (size_t)v2);
  const float2* mp2 = (const float2*)mp;         // B64 gathers
  float2 m0 = mp2[v0], m1 = mp2[v1], m2 = mp2[v2];

  float em[7], ei[7];
#pragma unroll
  for (int j = 0; j < 7; ++j) { em[j] = emean[j]; ei[j] = fast_rcp(estd[j]); }

  do_pair(eout, (size_t)C, (size_t)cid, 0, w0, w1, m0, m1, em, ei);
  do_pair(eout, (size_t)C, (size_t)cid, 1, w1, w2, m1, m2, em, ei);
  do_pair(eout, (size_t)C, (size_t)cid, 2, w2, w0, m2, m0, em, ei);
}

// ---------------------------------------------------------------------------
// Launch. d_in order: world_pos, prev_world_pos, node_type, mesh_pos, cells,
// node_mean, node_std, edge_mean, edge_std.
// d_out = node_features_norm (N*12) || edge_features_norm (6C*7), float32.
// ---------------------------------------------------------------------------
extern "C" void kernel_launch(void* const* d_in, const int* in_sizes, int n_in,
                              void* d_out, int out_size, void* d_ws, size_t ws_size,
                              hipStream_t stream) {
  const float* wp    = (const float*)d_in[0];
  const float* pwp   = (const float*)d_in[1];
  const int*   nt    = (const int*)d_in[2];
  const float* mp    = (const float*)d_in[3];
  const int*   cells = (const int*)d_in[4];
  const float* nmean = (const float*)d_in[5];
  const float* nstd  = (const float*)d_in[6];
  const float* emean = (const float*)d_in[7];
  const float* estd  = (const float*)d_in[8];

  const int N = in_sizes[0] / 3;   // 200,000
  const int C = in_sizes[4] / 3;   // 400,000

  float* out  = (float*)d_out;
  float* eout = out + (size_t)N * 12;

  node_feat_kernel<<<(N + 255) / 256, 256, 0, stream>>>(wp, pwp, nt, nmean, nstd, out, N);
  edge_feat_kernel<<<(C + 255) / 256, 256, 0, stream>>>(wp, mp, cells, emean, estd, eout, C, N);
}